// TERMinator2_9320079033225
// MI455X (gfx1250) — compile-verified
//
#include <hip/hip_runtime.h>
#include <stdint.h>
#include <math.h>

// Problem constants (fixed by the reference's setup_inputs).
#define NB 4
#define NL 2048
#define NK 30
#define NA 20
#define KP (NK - 1)            // 29 neighbor tiles used
#define TILE (NA * NA)         // 400 floats per tile
#define CHUNKS (KP * TILE / 4) // 2900 float4 chunks = 46400 bytes
#define LN_EPS 1e-5f

// ---- CDNA5 async global->LDS copy (tracked by ASYNCcnt) --------------------
#if defined(__has_builtin)
#if __has_builtin(__builtin_amdgcn_s_wait_asynccnt)
#define WAIT_ASYNCCNT0() __builtin_amdgcn_s_wait_asynccnt(0)
#endif
#endif
#ifndef WAIT_ASYNCCNT0
#define WAIT_ASYNCCNT0() asm volatile("s_wait_asynccnt 0x0" ::: "memory")
#endif

// GVS addressing: mem = SGPR64 base + VGPR32 offset; VDST = LDS byte address.
// etab is a one-touch 393 MB stream -> non-temporal hint keeps L2 for the
// heavily reused seqs/E_idx/self_etab data.
__device__ __forceinline__ void async_copy_b128_nt(uint32_t lds_byte_addr,
                                                   uint32_t voffset_bytes,
                                                   uint64_t global_base) {
  asm volatile("global_load_async_to_lds_b128 %0, %1, %2 th:TH_LOAD_NT"
               :: "v"(lds_byte_addr), "v"(voffset_bytes), "s"(global_base)
               : "memory");
}

// ---------------------------------------------------------------------------
// Stage 1: one (b,l) per single-wave32 block.
//   - async-stream 29 contiguous 20x20 f32 tiles (46.4 KB) into LDS (DMA runs
//     while the wave resolves the dependent E_idx -> seqs gather)
//   - gather column E_aa[k] of each tile, sum over k, add self energy
//   - LayerNorm(20) + log_softmax(20), select seqs[b,l], scale by mask
// ---------------------------------------------------------------------------
__global__ __launch_bounds__(32)
void terminator_nlpl_stage1(const float* __restrict__ self_etab,
                            const float* __restrict__ etab,
                            const int*   __restrict__ E_idx,
                            const int*   __restrict__ seqs,
                            const float* __restrict__ x_mask,
                            const float* __restrict__ ln_w,
                            const float* __restrict__ ln_b,
                            float* __restrict__ lp_out) {
  __shared__ float tile[KP * TILE];  // 46400 B
  __shared__ int   eaa[32];
  __shared__ float red[32];

  const int bl   = blockIdx.x;       // b*NL + l
  const int b    = bl / NL;
  const int lane = threadIdx.x;      // wave32

  // Kick off the DMA first: 2900 x 16B coalesced chunks, k=1..29 contiguous.
  const uint64_t gbase = (uint64_t)(uintptr_t)(etab + ((size_t)bl * NK + 1) * TILE);
  const uint32_t lbase = (uint32_t)(uintptr_t)(void*)&tile[0]; // low 32b = LDS offset
  for (int i = lane; i < CHUNKS; i += 32) {
    const uint32_t off = (uint32_t)i * 16u;
    async_copy_b128_nt(lbase + off, off, gbase);
  }

  // Overlap with the DMA: E_aa[k] = seqs[b, E_idx[b,l,k+1]]   (k = 0..28)
  if (lane < KP) {
    const int j = E_idx[(size_t)bl * NK + 1 + lane];
    eaa[lane] = seqs[b * NL + j];
  }

  WAIT_ASYNCCNT0();     // all LDS writes by the async engine complete
  __syncthreads();      // order eaa[] stores vs. reads below (DScnt + barrier)

  // Column gather + sum over k (stride 20 words -> only 2-way bank conflicts).
  float x = 0.0f;
  if (lane < NA) {
#pragma unroll 4
    for (int k = 0; k < KP; ++k)
      x += tile[k * TILE + lane * NA + eaa[k]];
    x += self_etab[(size_t)bl * NA + lane];
  }

  // ---- LayerNorm over the 20 lanes (broadcast reductions via LDS) ----
  red[lane] = (lane < NA) ? x : 0.0f;
  __syncthreads();
  float s = 0.0f;
  for (int a = 0; a < NA; ++a) s += red[a];
  const float mu = s * (1.0f / NA);
  const float d  = x - mu;
  __syncthreads();
  red[lane] = (lane < NA) ? d * d : 0.0f;
  __syncthreads();
  float vs = 0.0f;
  for (int a = 0; a < NA; ++a) vs += red[a];
  const float var = vs * (1.0f / NA);

  const float w  = (lane < NA) ? ln_w[lane] : 0.0f;
  const float bb = (lane < NA) ? ln_b[lane] : 0.0f;
  const float h  = d * rsqrtf(var + LN_EPS) * w + bb;

  // ---- log_softmax(-h) over the 20 lanes ----
  const float y = -h;
  __syncthreads();
  red[lane] = (lane < NA) ? y : -3.4e38f;
  __syncthreads();
  float m = -3.4e38f;
  for (int a = 0; a < NA; ++a) m = fmaxf(m, red[a]);
  const float e = (lane < NA) ? expf(y - m) : 0.0f;
  __syncthreads();
  red[lane] = e;
  __syncthreads();
  float se = 0.0f;
  for (int a = 0; a < NA; ++a) se += red[a];
  const float lse = m + logf(se);
  __syncthreads();
  red[lane] = y - lse;             // log-probabilities, lanes 0..19
  __syncthreads();

  if (lane == 0) {
    const int sel = seqs[bl];
    lp_out[bl] = red[sel] * x_mask[bl];
  }
}

// ---------------------------------------------------------------------------
// Stage 2: deterministic tree reduction -> scalar  -mean_b( sum_l lp / sum_l mask )
// ---------------------------------------------------------------------------
__global__ __launch_bounds__(256)
void terminator_nlpl_stage2(const float* __restrict__ lp,
                            const float* __restrict__ x_mask,
                            float* __restrict__ out) {
  __shared__ float ssum[256];
  __shared__ float smask[256];
  const int t = threadIdx.x;
  float acc = 0.0f;
  for (int b = 0; b < NB; ++b) {
    float s = 0.0f, mk = 0.0f;
    for (int i = t; i < NL; i += 256) {
      s  += lp[b * NL + i];
      mk += x_mask[b * NL + i];
    }
    ssum[t] = s; smask[t] = mk;
    __syncthreads();
    for (int off = 128; off > 0; off >>= 1) {
      if (t < off) { ssum[t] += ssum[t + off]; smask[t] += smask[t + off]; }
      __syncthreads();
    }
    if (t == 0) acc += ssum[0] / smask[0];
    __syncthreads();
  }
  if (t == 0) out[0] = -(acc / NB);
}

// ---------------------------------------------------------------------------
extern "C" void kernel_launch(void* const* d_in, const int* in_sizes, int n_in,
                              void* d_out, int out_size, void* d_ws, size_t ws_size,
                              hipStream_t stream) {
  (void)in_sizes; (void)n_in; (void)out_size; (void)ws_size;
  const float* self_etab = (const float*)d_in[0];
  const float* etab      = (const float*)d_in[1];
  const int*   E_idx     = (const int*)d_in[2];
  const int*   seqs      = (const int*)d_in[3];
  const float* x_mask    = (const float*)d_in[4];
  const float* ln_w      = (const float*)d_in[5];
  const float* ln_b      = (const float*)d_in[6];
  float* lp = (float*)d_ws;  // NB*NL floats = 32 KB scratch

  terminator_nlpl_stage1<<<NB * NL, 32, 0, stream>>>(
      self_etab, etab, E_idx, seqs, x_mask, ln_w, ln_b, lp);
  terminator_nlpl_stage2<<<1, 256, 0, stream>>>(lp, x_mask, (float*)d_out);
}